// VTMUpsampler_78735340470846
// MI455X (gfx1250) — compile-verified
//
#include <hip/hip_runtime.h>

// ---------------------------------------------------------------------------
// Fused VTM 2x upsampler for MI455X (gfx1250, wave32).
//   - memory-bound problem (~125 MB min traffic -> ~5.4us @ 23.3 TB/s), so we
//     fuse H+V passes: input staged once in LDS, intermediate never hits HBM.
//   - vertical pass done as D = Wv(16x16) x Hbuf(16x16) via 4 chained
//     V_WMMA_F32_16X16X4_F32 (full fp32, K=16 contraction).
// ---------------------------------------------------------------------------

typedef __attribute__((ext_vector_type(2))) float v2f;
typedef __attribute__((ext_vector_type(8))) float v8f;

#define H_IN   540
#define W_IN   960
#define H_OUT  1080
#define W_OUT  1920
#define LDSH   16    // 15 input rows needed per 16-output-row tile (+1 pad)
#define LDSW   40    // 39 input cols needed per 64-output-col tile (+1 pad)

__global__ __launch_bounds__(128)
void vtm_upsample2x_fused(const float* __restrict__ x,
                          const float* __restrict__ taps,   // [16][8] flat
                          float* __restrict__ out)
{
    __shared__ float lds[LDSH][LDSW];

    const int tx   = blockIdx.x;           // 0..29   : 64-output-col tiles
    const int ty   = blockIdx.y;           // 0..67   : 16-output-row tiles
    const int img  = blockIdx.z;           // 0..11   : B*C planes
    const int tid  = threadIdx.x;          // 0..127
    const int wave = tid >> 5;             // 0..3 (wave32)
    const int lane = tid & 31;
    const int hi   = lane >> 4;            // lane half (0/1)
    const int lm   = lane & 15;

    const int oy0 = ty * 16;
    const int ox0 = tx * 64;
    const int iy0 = (oy0 >> 1) - 3;        // first intermediate row of window
    const int ix0 = (ox0 >> 1) - 3;        // first input col of window

    const float* __restrict__ xin = x + (size_t)img * (H_IN * W_IN);

    // ---- 1) stage clamped input tile into LDS (16 x 40) --------------------
    for (int e = tid; e < LDSH * LDSW; e += 128) {
        int li = e / LDSW;
        int lj = e - li * LDSW;
        int gy = iy0 + li; gy = gy < 0 ? 0 : (gy >= H_IN ? H_IN - 1 : gy);
        int gx = ix0 + lj; gx = gx < 0 ? 0 : (gx >= W_IN ? W_IN - 1 : gx);
        lds[li][lj] = xin[(size_t)gy * W_IN + gx];
    }
    __syncthreads();

    // ---- 2) horizontal 8-tap pass directly into B-fragment layout ----------
    // this wave's 16-column slice of the 64-col tile
    const int oxg    = ox0 + wave * 16 + lm;     // global output column
    const int parity = oxg & 1;                  // scale=2 -> phase 0 or 8
    float wh[8];
#pragma unroll
    for (int k = 0; k < 8; ++k)
        wh[k] = taps[parity * 64 + k];           // taps[8*parity][k]

    const int j0 = wave * 8 + (lm >> 1);         // = oxg/2 - 3 - ix0

    // B layout (4x16 f32, wave32): B_t.x -> K = 4t + 2*hi, B_t.y -> K = 4t+1+2*hi,
    // N = lane&15. K here = local intermediate row index.
    v2f Bf[4];
#pragma unroll
    for (int t = 0; t < 4; ++t) {
#pragma unroll
        for (int p = 0; p < 2; ++p) {
            const int li = 4 * t + p + 2 * hi;
            float acc = 0.0f;
#pragma unroll
            for (int k = 0; k < 8; ++k)
                acc += wh[k] * lds[li][j0 + k];
            if (p == 0) Bf[t].x = acc; else Bf[t].y = acc;
        }
    }

    // ---- 3) constant banded vertical-weight matrix as A-fragments ----------
    // A layout (16x4 f32): M = lane&15, K = vgpr + 2*(lane/16).
    // Wv[m][kk] = taps[8*(m&1)][kk - m/2] if 0 <= kk - m/2 < 8 else 0.
    v2f Af[4];
#pragma unroll
    for (int t = 0; t < 4; ++t) {
#pragma unroll
        for (int p = 0; p < 2; ++p) {
            const int kk = 4 * t + p + 2 * hi;
            const int d  = kk - (lm >> 1);
            const float w = (d >= 0 && d < 8) ? taps[(lm & 1) * 64 + d] : 0.0f;
            if (p == 0) Af[t].x = w; else Af[t].y = w;
        }
    }

    // ---- 4) vertical pass: K=16 contraction via 4 chained f32 WMMAs --------
    v8f D = {0.f, 0.f, 0.f, 0.f, 0.f, 0.f, 0.f, 0.f};
#pragma unroll
    for (int t = 0; t < 4; ++t) {
        // 8 args: (neg_a, A, neg_b, B, c_mod, C, reuse_a, reuse_b)
        D = __builtin_amdgcn_wmma_f32_16x16x4_f32(
                false, Af[t], false, Bf[t], (short)0, D, false, false);
    }

    // ---- 5) fixed-point epilogue + guarded store ---------------------------
    // reference: (acc + 2048) / 4096 == acc * 2^-12 + 0.5 (exact, pow2)
    float* __restrict__ o = out + (size_t)img * ((size_t)H_OUT * W_OUT);
    const float inv = 1.0f / 4096.0f;
#pragma unroll
    for (int j = 0; j < 8; ++j) {
        const int row = oy0 + j + 8 * hi;        // C/D layout: M = j + 8*hi
        if (row < H_OUT)
            o[(size_t)row * W_OUT + oxg] = D[j] * inv + 0.5f;
    }
}

// ---------------------------------------------------------------------------
// Harness entry point.
//   d_in[0] = x     (4*3*540*960 f32)
//   d_in[1] = taps  (16*8 f32)
//   d_in[2] = scale (int scalar, == 2; phase math specialized for scale=2)
//   d_out   = (4*3*1080*1920 f32)
// ---------------------------------------------------------------------------
extern "C" void kernel_launch(void* const* d_in, const int* in_sizes, int n_in,
                              void* d_out, int out_size, void* d_ws, size_t ws_size,
                              hipStream_t stream) {
    (void)in_sizes; (void)n_in; (void)d_ws; (void)ws_size; (void)out_size;
    const float* x    = (const float*)d_in[0];
    const float* taps = (const float*)d_in[1];
    float*       out  = (float*)d_out;

    dim3 grid(W_OUT / 64,                 // 30 column tiles
              (H_OUT + 15) / 16,          // 68 row tiles
              12);                        // B*C planes
    dim3 block(128);                      // 4 wave32 waves
    vtm_upsample2x_fused<<<grid, block, 0, stream>>>(x, taps, out);
}